// SPDSRU_59940563583848
// MI455X (gfx1250) — compile-verified
//
#include <hip/hip_runtime.h>

// SPDSRU for MI455X (gfx1250, wave32). Memory-bound problem (~570 MB traffic,
// ~3.2 GFLOP): stream state/inputs once (state twice), do all six 32x32x32
// GEMMs per batch element with V_WMMA_F32_16X16X4_F32.

#define BATCH_N 8192
#define A_NUM   5
#define EPSF    1e-10f
#define WAVES   4

typedef __attribute__((ext_vector_type(2))) float v2f;
typedef __attribute__((ext_vector_type(8))) float v8f;

// ------------------------------------------------------------------
// Kernel 1: Cayley matrices  M = (I - S)^-1 (I + S),  S = skew(Bvec).
// One 32-thread wave per matrix (blockIdx selects Br / Bt / By).
// I - S has SPD symmetric part => Gauss-Jordan without pivoting is stable.
// ------------------------------------------------------------------
__global__ __launch_bounds__(32) void cayley_kernel(
    const float* __restrict__ Br, const float* __restrict__ Bt,
    const float* __restrict__ By, float* __restrict__ Mout)
{
  const float* Bv = (blockIdx.x == 0) ? Br : (blockIdx.x == 1) ? Bt : By;
  float* out = Mout + blockIdx.x * 1024;

  __shared__ float Am[32][33];
  __shared__ float Bm[32][33];
  const int t = threadIdx.x;  // row owner

  for (int c = 0; c < 32; ++c) {
    int i1 = t - 1 + c; i1 = i1 < 0 ? 0 : (i1 > 495 ? 495 : i1);
    int i2 = c - 1 + t; i2 = i2 < 0 ? 0 : (i2 > 495 ? 495 : i2);
    float Lrc = (c < t) ? Bv[i1] : 0.0f;  // lower-tri from Bvec
    float Lcr = (t < c) ? Bv[i2] : 0.0f;  // transpose element
    float S = Lrc - Lcr;                  // skew
    float I = (t == c) ? 1.0f : 0.0f;
    Am[t][c] = I - S;
    Bm[t][c] = I + S;
  }
  __syncthreads();

  for (int k = 0; k < 32; ++k) {
    float inv = 1.0f / Am[k][k];          // all lanes read pre-normalization
    __syncthreads();
    if (t == k)
      for (int j = 0; j < 32; ++j) { Am[k][j] *= inv; Bm[k][j] *= inv; }
    __syncthreads();
    if (t != k) {
      float f = Am[t][k];
      for (int j = 0; j < 32; ++j) { Am[t][j] -= f * Am[k][j]; Bm[t][j] -= f * Bm[k][j]; }
    }
    __syncthreads();
  }
  for (int c = 0; c < 32; ++c) out[t * 32 + c] = Bm[t][c];
}

// ------------------------------------------------------------------
// Per-wave 32x32x32 f32 GEMM via V_WMMA_F32_16X16X4_F32.
// D[m][n] = sum_k A[m][k] * (TRANSB ? B[n][k] : B[k][n]); epilogue gets (m,n,val).
// Fragment layouts per CDNA5 ISA (wave32):
//   A 16x4 : lane l -> A[m=l%16][k0 + 2*(l/16) + vgpr]
//   B 4x16 : lane l -> B[k0 + 2*(l/16) + vgpr][n=l%16]
//   C 16x16: lane l, vgpr v -> (m = v + 8*(l/16), n = l%16)
// ------------------------------------------------------------------
template <bool TRANSB, class EPI>
__device__ __forceinline__ void wave_gemm32(
    const float (&A)[32][33], const float (&B)[32][33], int lane, EPI epi)
{
  const int lh = lane >> 4;
  const int lm = lane & 15;
#pragma unroll
  for (int mi = 0; mi < 2; ++mi)
#pragma unroll
    for (int nj = 0; nj < 2; ++nj) {
      v8f acc = {0.f, 0.f, 0.f, 0.f, 0.f, 0.f, 0.f, 0.f};
#pragma unroll
      for (int k0 = 0; k0 < 32; k0 += 4) {
        const int ka = k0 + 2 * lh;
        v2f af, bf;
        af.x = A[mi * 16 + lm][ka + 0];
        af.y = A[mi * 16 + lm][ka + 1];
        if constexpr (TRANSB) {
          bf.x = B[nj * 16 + lm][ka + 0];
          bf.y = B[nj * 16 + lm][ka + 1];
        } else {
          bf.x = B[ka + 0][nj * 16 + lm];
          bf.y = B[ka + 1][nj * 16 + lm];
        }
        acc = __builtin_amdgcn_wmma_f32_16x16x4_f32(
            false, af, false, bf, (short)0, acc, false, false);
      }
#pragma unroll
      for (int v = 0; v < 8; ++v)
        epi(mi * 16 + v + 8 * lh, nj * 16 + lm, acc[v]);
    }
}

struct WaveBuf { float b0[32][33]; float b1[32][33]; float bx[32][33]; };

// ------------------------------------------------------------------
// Kernel 2: one wave (32 lanes) per batch element; 4 waves per block.
// ------------------------------------------------------------------
__global__ __launch_bounds__(WAVES * 32) void spdsru_kernel(
    const float* __restrict__ X, const float* __restrict__ state,
    const float* __restrict__ WRr, const float* __restrict__ Wtr,
    const float* __restrict__ Wphir, const float* __restrict__ Wsr,
    const float* __restrict__ Mmats, float* __restrict__ out)
{
  __shared__ float MR[32][33], MT[32][33], MY[32][33];
  __shared__ WaveBuf wb[WAVES];

  const int tid = threadIdx.x;
  for (int i = tid; i < 1024; i += WAVES * 32) {
    int r = i >> 5, c = i & 31;
    MR[r][c] = Mmats[i];
    MT[r][c] = Mmats[1024 + i];
    MY[r][c] = Mmats[2048 + i];
  }
  __syncthreads();

  // Uniform scalar weights (scalar loads; trivially cheap).
  const float ALPHA[A_NUM] = {0.01f, 0.25f, 0.5f, 0.9f, 0.99f};
  float w[A_NUM], q[A_NUM];
  float sw = 0.f, ss = 0.f;
#pragma unroll
  for (int a = 0; a < A_NUM; ++a) { float v = WRr[a]; w[a] = v * v; sw += v * v; }
#pragma unroll
  for (int a = 0; a < A_NUM; ++a) w[a] = w[a] / (sw + EPSF);
  const float wt2 = Wtr[0] * Wtr[0];
  const float wp2 = Wphir[0] * Wphir[0];
  const float at_c = wt2 / (wt2 + wp2 + EPSF);
  const float omat = 1.0f - at_c;
  float cphi = 0.f;
#pragma unroll
  for (int a = 0; a < A_NUM; ++a) { float v = Wsr[a]; q[a] = v * v; ss += v * v; }
#pragma unroll
  for (int a = 0; a < A_NUM; ++a) {
    float s = q[a] / (ss + EPSF);
    q[a] = s * (1.0f - ALPHA[a]);   // coefficient of Mt_1[a] in St
    cphi += s * ALPHA[a];           // coefficient of Phit in St
  }

  const int wave = tid >> 5;
  const int lane = tid & 31;
  const int b = blockIdx.x * WAVES + wave;
  WaveBuf& Wb = wb[wave];

  const float* __restrict__ stb = state + (size_t)b * (A_NUM * 1024);
  const float* __restrict__ xb  = X + (size_t)b * 1024;

  // ---- Pass 1: Yt = sum_a w_a * Mt_1[a] -> b0 ;  X -> bx  (float4 streams)
  float yt[32];
#pragma unroll
  for (int i = 0; i < 32; ++i) yt[i] = 0.f;
#pragma unroll
  for (int a = 0; a < A_NUM; ++a) {
    const float4* m4 = (const float4*)(stb + a * 1024);
    const float wa = w[a];
#pragma unroll
    for (int ch = 0; ch < 8; ++ch) {
      float4 m = m4[ch * 32 + lane];
      yt[ch * 4 + 0] += wa * m.x; yt[ch * 4 + 1] += wa * m.y;
      yt[ch * 4 + 2] += wa * m.z; yt[ch * 4 + 3] += wa * m.w;
    }
  }
  {
    const float4* x4 = (const float4*)xb;
#pragma unroll
    for (int ch = 0; ch < 8; ++ch) {
      float4 xv = x4[ch * 32 + lane];
      const int row = 4 * ch + (lane >> 3);
      const int col = 4 * (lane & 7);
      Wb.b0[row][col + 0] = yt[ch * 4 + 0]; Wb.b0[row][col + 1] = yt[ch * 4 + 1];
      Wb.b0[row][col + 2] = yt[ch * 4 + 2]; Wb.b0[row][col + 3] = yt[ch * 4 + 3];
      Wb.bx[row][col + 0] = xv.x; Wb.bx[row][col + 1] = xv.y;
      Wb.bx[row][col + 2] = xv.z; Wb.bx[row][col + 3] = xv.w;
    }
  }

  // ---- Rt = MR * Yt * MR^T ;  Tt = (1-at)*X + at*Rt  (into b0)
  wave_gemm32<false>(MR, Wb.b0, lane,
      [&](int r, int c, float v) { Wb.b1[r][c] = v; });
  wave_gemm32<true>(Wb.b1, MR, lane,
      [&](int r, int c, float v) { Wb.b0[r][c] = omat * Wb.bx[r][c] + at_c * v; });

  // ---- Phit = MT * Tt * MT^T  (into b0)
  wave_gemm32<false>(MT, Wb.b0, lane,
      [&](int r, int c, float v) { Wb.b1[r][c] = v; });
  wave_gemm32<true>(Wb.b1, MT, lane,
      [&](int r, int c, float v) { Wb.b0[r][c] = v; });

  // ---- Restream Mt_1: write Mt[a] = (1-alpha)Mt_1 + alpha*Phit,
  //      accumulate Z = sum_a q_a * Mt_1[a]  (St = Z + cphi*Phit)
  float z[32];
#pragma unroll
  for (int i = 0; i < 32; ++i) z[i] = 0.f;
  float* __restrict__ outMt = out + (size_t)BATCH_N * 1024 + (size_t)b * (A_NUM * 1024);
#pragma unroll
  for (int a = 0; a < A_NUM; ++a) {
    const float4* m4 = (const float4*)(stb + a * 1024);
    float4* o4 = (float4*)(outMt + a * 1024);
    const float al = ALPHA[a], oma = 1.0f - ALPHA[a], qa = q[a];
#pragma unroll
    for (int ch = 0; ch < 8; ++ch) {
      float4 m = m4[ch * 32 + lane];
      const int row = 4 * ch + (lane >> 3);
      const int col = 4 * (lane & 7);
      float4 o;
      o.x = oma * m.x + al * Wb.b0[row][col + 0];
      o.y = oma * m.y + al * Wb.b0[row][col + 1];
      o.z = oma * m.z + al * Wb.b0[row][col + 2];
      o.w = oma * m.w + al * Wb.b0[row][col + 3];
      z[ch * 4 + 0] += qa * m.x; z[ch * 4 + 1] += qa * m.y;
      z[ch * 4 + 2] += qa * m.z; z[ch * 4 + 3] += qa * m.w;
      o4[ch * 32 + lane] = o;
    }
  }
#pragma unroll
  for (int ch = 0; ch < 8; ++ch) {   // St -> bx
    const int row = 4 * ch + (lane >> 3);
    const int col = 4 * (lane & 7);
    Wb.bx[row][col + 0] = z[ch * 4 + 0] + cphi * Wb.b0[row][col + 0];
    Wb.bx[row][col + 1] = z[ch * 4 + 1] + cphi * Wb.b0[row][col + 1];
    Wb.bx[row][col + 2] = z[ch * 4 + 2] + cphi * Wb.b0[row][col + 2];
    Wb.bx[row][col + 3] = z[ch * 4 + 3] + cphi * Wb.b0[row][col + 3];
  }

  // ---- Ot = MY * St * MY^T -> b0, then coalesced float4 store
  wave_gemm32<false>(MY, Wb.bx, lane,
      [&](int r, int c, float v) { Wb.b1[r][c] = v; });
  wave_gemm32<true>(Wb.b1, MY, lane,
      [&](int r, int c, float v) { Wb.b0[r][c] = v; });

  float4* oOt = (float4*)(out + (size_t)b * 1024);
#pragma unroll
  for (int ch = 0; ch < 8; ++ch) {
    const int row = 4 * ch + (lane >> 3);
    const int col = 4 * (lane & 7);
    float4 o = { Wb.b0[row][col + 0], Wb.b0[row][col + 1],
                 Wb.b0[row][col + 2], Wb.b0[row][col + 3] };
    oOt[ch * 32 + lane] = o;
  }
}

extern "C" void kernel_launch(void* const* d_in, const int* in_sizes, int n_in,
                              void* d_out, int out_size, void* d_ws, size_t ws_size,
                              hipStream_t stream) {
  (void)in_sizes; (void)n_in; (void)out_size; (void)ws_size;
  const float* X     = (const float*)d_in[0];
  const float* state = (const float*)d_in[1];
  const float* WRr   = (const float*)d_in[2];
  const float* Wtr   = (const float*)d_in[3];
  const float* Wphir = (const float*)d_in[4];
  const float* Wsr   = (const float*)d_in[5];
  const float* Br    = (const float*)d_in[6];
  const float* Bt    = (const float*)d_in[7];
  const float* By    = (const float*)d_in[8];
  float* out  = (float*)d_out;
  float* mats = (float*)d_ws;  // 3 * 1024 floats: M_R, M_T, M_Y

  cayley_kernel<<<3, 32, 0, stream>>>(Br, Bt, By, mats);
  spdsru_kernel<<<BATCH_N / WAVES, WAVES * 32, 0, stream>>>(
      X, state, WRr, Wtr, Wphir, Wsr, mats, out);
}